// MultiScaleFeatureFusion_88278757802661
// MI455X (gfx1250) — compile-verified
//
#include <hip/hip_runtime.h>
#include <hip/hip_bf16.h>
#include <math.h>

// Problem constants from the reference
#define NN 50000
#define DD 128
#define EE 800000
#define NHOP 3

typedef __attribute__((ext_vector_type(2))) float v2f;
typedef __attribute__((ext_vector_type(8))) float v8f;

// ---------------------------------------------------------------------------
// 1) max(node_depths) reduction + scale-gate MLP (2 -> 32 -> 3, sigmoid)
// ---------------------------------------------------------------------------
__global__ void gate_kernel(const int* __restrict__ depths, int n,
                            const float* __restrict__ Wg1, const float* __restrict__ bg1,
                            const float* __restrict__ Wg2, const float* __restrict__ bg2,
                            float* __restrict__ sw) {
    __shared__ int smax[256];
    int tid = threadIdx.x;
    int m = 0;
    for (int i = tid; i < n; i += 256) m = max(m, depths[i]);
    smax[tid] = m;
    __syncthreads();
    for (int s = 128; s > 0; s >>= 1) {
        if (tid < s) smax[tid] = max(smax[tid], smax[tid + s]);
        __syncthreads();
    }
    if (tid == 0) {
        float sf0 = (float)n / 5000.0f;
        float sf1 = (float)smax[0] / 20.0f;
        float a0 = bg2[0], a1 = bg2[1], a2 = bg2[2];
        for (int j = 0; j < DD / 4; ++j) {
            float h = sf0 * Wg1[j] + sf1 * Wg1[DD / 4 + j] + bg1[j];
            h = fmaxf(h, 0.0f);
            a0 += h * Wg2[j * NHOP + 0];
            a1 += h * Wg2[j * NHOP + 1];
            a2 += h * Wg2[j * NHOP + 2];
        }
        sw[0] = 1.0f / (1.0f + __expf(-a0));
        sw[1] = 1.0f / (1.0f + __expf(-a1));
        sw[2] = 1.0f / (1.0f + __expf(-a2));
    }
}

// ---------------------------------------------------------------------------
// 2) Degree counts (shared by both scatter_mean rounds)
// ---------------------------------------------------------------------------
__global__ void degree_kernel(const int* __restrict__ col, float* __restrict__ counts, int e) {
    int i = blockIdx.x * blockDim.x + threadIdx.x;
    if (i < e) unsafeAtomicAdd(&counts[col[i]], 1.0f);
}

// ---------------------------------------------------------------------------
// 3) scatter-add: agg[col[e], :] += x[row[e], :]   (float4 per thread)
// ---------------------------------------------------------------------------
__global__ void scatter_add_kernel(const float* __restrict__ x,
                                   const int* __restrict__ row, const int* __restrict__ col,
                                   float* __restrict__ agg, int e) {
    long gid = (long)blockIdx.x * blockDim.x + threadIdx.x;
    long total = (long)e * (DD / 4);
    if (gid >= total) return;
    int ed = (int)(gid >> 5);            // DD/4 == 32 chunks per edge
    int c4 = (int)(gid & 31) * 4;
    int r = row[ed], c = col[ed];
    const float4 v = *(const float4*)(x + (long)r * DD + c4);
    float* dst = agg + (long)c * DD + c4;
    unsafeAtomicAdd(dst + 0, v.x);
    unsafeAtomicAdd(dst + 1, v.y);
    unsafeAtomicAdd(dst + 2, v.z);
    unsafeAtomicAdd(dst + 3, v.w);
}

// ---------------------------------------------------------------------------
// 4) divide by clamped degree
// ---------------------------------------------------------------------------
__global__ void divide_kernel(float* __restrict__ agg, const float* __restrict__ counts, int n) {
    long gid = (long)blockIdx.x * blockDim.x + threadIdx.x;
    long total = (long)n * (DD / 4);
    if (gid >= total) return;
    int node = (int)(gid >> 5);
    int off = (int)(gid & 31) * 4;
    float inv = 1.0f / fmaxf(counts[node], 1.0f);
    float4 v = *(float4*)(agg + (long)node * DD + off);
    v.x *= inv; v.y *= inv; v.z *= inv; v.w *= inv;
    *(float4*)(agg + (long)node * DD + off) = v;
}

// ---------------------------------------------------------------------------
// 5) Hop GEMM via V_WMMA_F32_16X16X4_F32:  Y = (X @ W + b) * sw[hop]
//    One wave per 16x16 tile; 8 waves/block cover 128 output cols.
// ---------------------------------------------------------------------------
__global__ void __launch_bounds__(256)
gemm_hop_kernel(const float* __restrict__ X, const float* __restrict__ W,
                const float* __restrict__ b, const float* __restrict__ sw, int hop,
                float* __restrict__ Y) {
    const int row0 = blockIdx.x * 16;
    const int wave = threadIdx.x >> 5;
    const int col0 = wave * 16;
    const int lane = threadIdx.x & 31;
    const int half = lane >> 4;
    const int l    = lane & 15;

    v8f acc = {};
    const float* xr = X + (long)(row0 + l) * DD + 2 * half;   // A row for this lane
    const float* wc = W + (long)(2 * half) * DD + col0 + l;   // B col for this lane
#pragma unroll
    for (int k = 0; k < DD; k += 4) {
        v2f a, bb;
        a.x = xr[k];
        a.y = xr[k + 1];
        bb.x = wc[(long)k * DD];
        bb.y = wc[(long)(k + 1) * DD];
        acc = __builtin_amdgcn_wmma_f32_16x16x4_f32(false, a, false, bb,
                                                    (short)0, acc, false, false);
    }
    const float s = sw[hop];
    const float bias = b[col0 + l];
#pragma unroll
    for (int j = 0; j < 8; ++j) {
        int m = j + 8 * half;
        Y[(long)(row0 + m) * DD + col0 + l] = (acc[j] + bias) * s;
    }
}

// ---------------------------------------------------------------------------
// 6) Attention: softmax(concat(T0,T1,T2) @ Wa + ba) -> att[N,3]
// ---------------------------------------------------------------------------
__global__ void attention_kernel(const float* __restrict__ T0, const float* __restrict__ T1,
                                 const float* __restrict__ T2,
                                 const float* __restrict__ Wa, const float* __restrict__ ba,
                                 float* __restrict__ att, int n) {
    int node = blockIdx.x * blockDim.x + threadIdx.x;
    if (node >= n) return;
    float l0 = ba[0], l1 = ba[1], l2 = ba[2];
    const float* Ts[NHOP] = {T0, T1, T2};
#pragma unroll
    for (int hop = 0; hop < NHOP; ++hop) {
        const float* t = Ts[hop] + (long)node * DD;
        const float* wa = Wa + (long)hop * DD * NHOP;
        for (int kk = 0; kk < DD; ++kk) {
            float xv = t[kk];
            l0 += xv * wa[kk * NHOP + 0];
            l1 += xv * wa[kk * NHOP + 1];
            l2 += xv * wa[kk * NHOP + 2];
        }
    }
    float mx = fmaxf(l0, fmaxf(l1, l2));
    float e0 = __expf(l0 - mx), e1 = __expf(l1 - mx), e2 = __expf(l2 - mx);
    float inv = 1.0f / (e0 + e1 + e2);
    att[(long)node * NHOP + 0] = e0 * inv;
    att[(long)node * NHOP + 1] = e1 * inv;
    att[(long)node * NHOP + 2] = e2 * inv;
}

// ---------------------------------------------------------------------------
// 7) Fusion GEMM (K = 384) with per-node/per-hop attention folded into A-frag:
//    out = concat_i(T_i * att_i) @ Wf + bf
// ---------------------------------------------------------------------------
__global__ void __launch_bounds__(256)
fusion_kernel(const float* __restrict__ T0, const float* __restrict__ T1,
              const float* __restrict__ T2, const float* __restrict__ att,
              const float* __restrict__ Wf, const float* __restrict__ bf,
              float* __restrict__ out) {
    const int row0 = blockIdx.x * 16;
    const int wave = threadIdx.x >> 5;
    const int col0 = wave * 16;
    const int lane = threadIdx.x & 31;
    const int half = lane >> 4;
    const int l    = lane & 15;

    const int row = row0 + l;                 // A-matrix row owned by this lane
    const float a_att0 = att[(long)row * NHOP + 0];
    const float a_att1 = att[(long)row * NHOP + 1];
    const float a_att2 = att[(long)row * NHOP + 2];
    const float* Ts[NHOP] = {T0, T1, T2};

    v8f acc = {};
#pragma unroll
    for (int k = 0; k < NHOP * DD; k += 4) {
        const int hop = k >> 7;
        const float amul = (hop == 0) ? a_att0 : ((hop == 1) ? a_att1 : a_att2);
        const float* t = Ts[hop] + (long)row * DD + (k & (DD - 1)) + 2 * half;
        const float* wf = Wf + (long)(k + 2 * half) * DD + col0 + l;
        v2f a, bb;
        a.x = t[0] * amul;
        a.y = t[1] * amul;
        bb.x = wf[0];
        bb.y = wf[DD];
        acc = __builtin_amdgcn_wmma_f32_16x16x4_f32(false, a, false, bb,
                                                    (short)0, acc, false, false);
    }
    const float bias = bf[col0 + l];
#pragma unroll
    for (int j = 0; j < 8; ++j) {
        int m = j + 8 * half;
        out[(long)(row0 + m) * DD + col0 + l] = acc[j] + bias;
    }
}

// ---------------------------------------------------------------------------
// Launch
// ---------------------------------------------------------------------------
static inline int cdiv(long a, long b) { return (int)((a + b - 1) / b); }

extern "C" void kernel_launch(void* const* d_in, const int* in_sizes, int n_in,
                              void* d_out, int out_size, void* d_ws, size_t ws_size,
                              hipStream_t stream) {
    const float* x      = (const float*)d_in[0];   // [N, D]
    const int*   ei     = (const int*)  d_in[1];   // [2, E]
    const int*   depths = (const int*)  d_in[2];   // [N]
    const float* Wh     = (const float*)d_in[3];   // [3, D, D]
    const float* bh     = (const float*)d_in[4];   // [3, D]
    const float* Wg1    = (const float*)d_in[5];
    const float* bg1    = (const float*)d_in[6];
    const float* Wg2    = (const float*)d_in[7];
    const float* bg2    = (const float*)d_in[8];
    const float* Wa     = (const float*)d_in[9];   // [3*D, 3]
    const float* ba     = (const float*)d_in[10];
    const float* Wf     = (const float*)d_in[11];  // [3*D, D]
    const float* bf     = (const float*)d_in[12];
    float* out = (float*)d_out;

    const int* row = ei;
    const int* col = ei + EE;

    // Workspace carve-out (256B aligned)
    char* ws = (char*)d_ws;
    size_t o = 0;
    auto carve = [&](size_t bytes) -> float* {
        float* p = (float*)(ws + o);
        o += (bytes + 255) & ~(size_t)255;
        return p;
    };
    float* sw     = carve(16 * sizeof(float));
    float* counts = carve((size_t)NN * sizeof(float));
    float* agg1   = carve((size_t)NN * DD * sizeof(float));
    float* agg2   = carve((size_t)NN * DD * sizeof(float));
    float* T0     = carve((size_t)NN * DD * sizeof(float));
    float* T1     = carve((size_t)NN * DD * sizeof(float));
    float* T2     = carve((size_t)NN * DD * sizeof(float));
    float* att    = carve((size_t)NN * NHOP * sizeof(float));
    (void)ws_size; (void)in_sizes; (void)n_in; (void)out_size;

    hipMemsetAsync(counts, 0, (size_t)NN * sizeof(float), stream);
    hipMemsetAsync(agg1,   0, (size_t)NN * DD * sizeof(float), stream);
    hipMemsetAsync(agg2,   0, (size_t)NN * DD * sizeof(float), stream);

    // Scale gate
    gate_kernel<<<1, 256, 0, stream>>>(depths, NN, Wg1, bg1, Wg2, bg2, sw);

    // Degrees (shared by both scatter_mean rounds)
    degree_kernel<<<cdiv(EE, 256), 256, 0, stream>>>(col, counts, EE);

    // agg1 = scatter_mean(x)
    scatter_add_kernel<<<cdiv((long)EE * (DD / 4), 256), 256, 0, stream>>>(x, row, col, agg1, EE);
    divide_kernel<<<cdiv((long)NN * (DD / 4), 256), 256, 0, stream>>>(agg1, counts, NN);

    // agg2 = scatter_mean(agg1)
    scatter_add_kernel<<<cdiv((long)EE * (DD / 4), 256), 256, 0, stream>>>(agg1, row, col, agg2, EE);
    divide_kernel<<<cdiv((long)NN * (DD / 4), 256), 256, 0, stream>>>(agg2, counts, NN);

    // Hop transforms (WMMA f32 16x16x4), N = 3125 * 16 exactly
    const int rowTiles = NN / 16;
    gemm_hop_kernel<<<rowTiles, 256, 0, stream>>>(x,    Wh + 0 * DD * DD, bh + 0 * DD, sw, 0, T0);
    gemm_hop_kernel<<<rowTiles, 256, 0, stream>>>(agg1, Wh + 1 * DD * DD, bh + 1 * DD, sw, 1, T1);
    gemm_hop_kernel<<<rowTiles, 256, 0, stream>>>(agg2, Wh + 2 * DD * DD, bh + 2 * DD, sw, 2, T2);

    // Attention softmax over hops
    attention_kernel<<<cdiv(NN, 256), 256, 0, stream>>>(T0, T1, T2, Wa, ba, att, NN);

    // Fused output GEMM (WMMA, K = 384, attention folded into A-fragments)
    fusion_kernel<<<rowTiles, 256, 0, stream>>>(T0, T1, T2, att, Wf, bf, out);
}